// ScaleSpaceAffinePatchExtractor_80711025426538
// MI455X (gfx1250) — compile-verified
//
#include <hip/hip_runtime.h>
#include <cmath>
#include <algorithm>

#define NLEVELS 8
#define BORDER  16
#define KNUM    2000
#define PS      32
#define NOCT    3

typedef __attribute__((ext_vector_type(2))) float v2f;
typedef __attribute__((ext_vector_type(8))) float v8f;
typedef unsigned long long u64;

struct Taps { float t[16]; int r; };
struct Info {
  unsigned long long offF[3];   // float offset of each octave's 8-level slab in pyramid buffer
  unsigned long long px[3];     // pixels per level per octave
  int H[3]; int W[3];
  float sigma[NLEVELS];
};

// ---------------------------------------------------------------------------
// Separable Gaussian blur via V_WMMA_F32_16X16X4_F32, radius templated so the
// K-step loop fully unrolls (STEPS is 6 or 7) into one clustered load+WMMA
// stream. Taps live in a zero-padded LDS table at offset 16 so band-matrix
// fragments are a single unconditional ds_load (index always in [1,44)).
// ---------------------------------------------------------------------------

// Row pass: out[m,n] = sum_j src[row0+m, col0+j-r] * taps[j-n]
template <int R>
__global__ __launch_bounds__(256) void row_blur_wmma(const float* __restrict__ src,
                                                     float* __restrict__ dst,
                                                     int H, int W, Taps tp) {
  constexpr int K = 2 * R + 1;
  constexpr int STEPS = (16 + K + 2) >> 2;      // ceil((16+K-1)/4)
  __shared__ float st[48];
  if (threadIdx.x < 48)
    st[threadIdx.x] = (threadIdx.x >= 16 && threadIdx.x < 32) ? tp.t[threadIdx.x - 16] : 0.f;
  __syncthreads();
  const int wave = threadIdx.x >> 5;
  const int lane = threadIdx.x & 31;
  const int tilesX = (W + 15) >> 4;
  const int tilesY = (H + 15) >> 4;
  const int tile = blockIdx.x * 8 + wave;        // wave-uniform
  if (tile >= tilesX * tilesY) return;
  const int row0 = (tile / tilesX) << 4;
  const int col0 = (tile % tilesX) << 4;
  const int m    = lane & 15;                    // A-row / B-col / C-col for this lane
  const int half = lane >> 4;
  const int arow = row0 + m;
  v8f acc = {0.f, 0.f, 0.f, 0.f, 0.f, 0.f, 0.f, 0.f};

  const bool interior = (row0 + 15 < H) && (col0 - R >= 0) && (col0 + 4 * STEPS - 1 - R < W);
  if (interior) {
    const float* rp = src + (size_t)arow * W + (col0 - R);
#pragma unroll
    for (int s = 0; s < STEPS; ++s) {
      const int j0 = 4 * s + half * 2;
      v2f a; a.x = rp[j0]; a.y = rp[j0 + 1];
      v2f b; b.x = st[16 + j0 - m]; b.y = st[17 + j0 - m];
      acc = __builtin_amdgcn_wmma_f32_16x16x4_f32(false, a, false, b, (short)0, acc, false, false);
    }
    float* wp0 = dst + (size_t)(row0 + 8 * half) * W + (col0 + m);
#pragma unroll
    for (int v = 0; v < 8; ++v) wp0[(size_t)v * W] = acc[v];
  } else {
    const bool rok = arow < H;
    const float* rp = src + (size_t)min(arow, H - 1) * W;
#pragma unroll
    for (int s = 0; s < STEPS; ++s) {
      const int j0 = 4 * s + half * 2;
      const int c0 = col0 + j0 - R;
      const float ax = rp[min(max(c0, 0), W - 1)];
      const float ay = rp[min(max(c0 + 1, 0), W - 1)];
      v2f a;
      a.x = (rok && (unsigned)c0 < (unsigned)W) ? ax : 0.f;
      a.y = (rok && (unsigned)(c0 + 1) < (unsigned)W) ? ay : 0.f;
      v2f b; b.x = st[16 + j0 - m]; b.y = st[17 + j0 - m];
      acc = __builtin_amdgcn_wmma_f32_16x16x4_f32(false, a, false, b, (short)0, acc, false, false);
    }
    const int cc = col0 + m;
    if (cc < W) {
#pragma unroll
      for (int v = 0; v < 8; ++v) {
        const int rr = row0 + v + 8 * half;
        if (rr < H) dst[(size_t)rr * W + cc] = acc[v];
      }
    }
  }
}

// Column pass: out[m,n] = sum_j taps[j-m] * src[row0+j-r, col0+n]
template <int R>
__global__ __launch_bounds__(256) void col_blur_wmma(const float* __restrict__ src,
                                                     float* __restrict__ dst,
                                                     int H, int W, Taps tp) {
  constexpr int K = 2 * R + 1;
  constexpr int STEPS = (16 + K + 2) >> 2;
  __shared__ float st[48];
  if (threadIdx.x < 48)
    st[threadIdx.x] = (threadIdx.x >= 16 && threadIdx.x < 32) ? tp.t[threadIdx.x - 16] : 0.f;
  __syncthreads();
  const int wave = threadIdx.x >> 5;
  const int lane = threadIdx.x & 31;
  const int tilesX = (W + 15) >> 4;
  const int tilesY = (H + 15) >> 4;
  const int tile = blockIdx.x * 8 + wave;
  if (tile >= tilesX * tilesY) return;
  const int row0 = (tile / tilesX) << 4;
  const int col0 = (tile % tilesX) << 4;
  const int m    = lane & 15;
  const int half = lane >> 4;
  const int bcol = col0 + m;
  v8f acc = {0.f, 0.f, 0.f, 0.f, 0.f, 0.f, 0.f, 0.f};

  const bool interior = (col0 + 15 < W) && (row0 - R >= 0) && (row0 + 4 * STEPS - 1 - R < H);
  if (interior) {
    const float* cp = src + (size_t)(row0 - R) * W + bcol;
#pragma unroll
    for (int s = 0; s < STEPS; ++s) {
      const int j0 = 4 * s + half * 2;
      v2f a; a.x = st[16 + j0 - m]; a.y = st[17 + j0 - m];
      v2f b; b.x = cp[(size_t)j0 * W]; b.y = cp[(size_t)(j0 + 1) * W];
      acc = __builtin_amdgcn_wmma_f32_16x16x4_f32(false, a, false, b, (short)0, acc, false, false);
    }
    float* wp0 = dst + (size_t)(row0 + 8 * half) * W + bcol;
#pragma unroll
    for (int v = 0; v < 8; ++v) wp0[(size_t)v * W] = acc[v];
  } else {
    const bool cok = bcol < W;
    const int bc = min(bcol, W - 1);
#pragma unroll
    for (int s = 0; s < STEPS; ++s) {
      const int j0 = 4 * s + half * 2;
      v2f a; a.x = st[16 + j0 - m]; a.y = st[17 + j0 - m];
      const int r0 = row0 + j0 - R;
      const float bx = src[(size_t)min(max(r0, 0), H - 1) * W + bc];
      const float by = src[(size_t)min(max(r0 + 1, 0), H - 1) * W + bc];
      v2f b;
      b.x = (cok && (unsigned)r0 < (unsigned)H) ? bx : 0.f;
      b.y = (cok && (unsigned)(r0 + 1) < (unsigned)H) ? by : 0.f;
      acc = __builtin_amdgcn_wmma_f32_16x16x4_f32(false, a, false, b, (short)0, acc, false, false);
    }
    if (cok) {
#pragma unroll
      for (int v = 0; v < 8; ++v) {
        const int rr = row0 + v + 8 * half;
        if (rr < H) dst[(size_t)rr * W + bcol] = acc[v];
      }
    }
  }
}

// ---------------------------------------------------------------------------
__global__ void downsample_k(const float* __restrict__ src, float* __restrict__ dst,
                             int Hd, int Wd, int Ws) {
  const int x = blockIdx.x * blockDim.x + threadIdx.x;
  const int y = blockIdx.y;
  if (x < Wd) dst[(size_t)y * Wd + x] = src[(size_t)(2 * y) * Ws + 2 * x];
}

__global__ void hessian_k(const float* __restrict__ src, float* __restrict__ dst,
                          int H, int W, float s4) {
  const int x = blockIdx.x * blockDim.x + threadIdx.x;
  const int y = blockIdx.y;
  if (x >= W) return;
  float gxx, gyy, gxy, c;
  if (x > 0 && x < W - 1 && y > 0 && y < H - 1) {
    const float* p = src + (size_t)y * W + x;
    c = p[0];
    gxx = p[-1] - 2.f * c + p[1];
    gyy = p[-W] - 2.f * c + p[W];
    gxy = 0.25f * (p[-W - 1] - p[-W + 1] - p[W - 1] + p[W + 1]);
  } else {
    auto at = [&](int yy, int xx) -> float {
      return (yy >= 0 && yy < H && xx >= 0 && xx < W) ? src[(size_t)yy * W + xx] : 0.f;
    };
    c = at(y, x);
    gxx = at(y, x - 1) - 2.f * c + at(y, x + 1);
    gyy = at(y - 1, x) - 2.f * c + at(y + 1, x);
    gxy = 0.25f * (at(y - 1, x - 1) - at(y - 1, x + 1) - at(y + 1, x - 1) + at(y + 1, x + 1));
  }
  dst[(size_t)y * W + x] = (gxx * gyy - gxy * gxy) * s4;
}

// 3x3x3 scale-space NMS test (y,x guaranteed >= BORDER from edges)
__device__ __forceinline__ bool is_peak(const float* __restrict__ lo, const float* __restrict__ cu,
                                        const float* __restrict__ hi, int W, int y, int x, float c) {
#pragma unroll
  for (int dy = -1; dy <= 1; ++dy) {
    size_t b = (size_t)(y + dy) * W + x;
#pragma unroll
    for (int dx = -1; dx <= 1; ++dx) {
      if (c < cu[b + dx]) return false;     // cur >= maxpool3(cur)
      if (!(c > lo[b + dx])) return false;  // cur >  maxpool3(low)
      if (!(c > hi[b + dx])) return false;  // cur >  maxpool3(high)
    }
  }
  return true;
}

__global__ void nms_hist_k(const float* __restrict__ lo, const float* __restrict__ cu,
                           const float* __restrict__ hi, int H, int W, unsigned* __restrict__ hist) {
  __shared__ unsigned h[4096];
  for (int i = threadIdx.x; i < 4096; i += blockDim.x) h[i] = 0;
  __syncthreads();
  const int x = BORDER + blockIdx.x * blockDim.x + threadIdx.x;
  const int y = BORDER + blockIdx.y;
  if (x < W - BORDER) {
    const float c = cu[(size_t)y * W + x];
    if (c > 0.f && is_peak(lo, cu, hi, W, y, x, c))
      atomicAdd(&h[__float_as_uint(c) >> 20], 1u);  // positive floats: bits are order-monotone
  }
  __syncthreads();
  for (int i = threadIdx.x; i < 4096; i += blockDim.x)
    if (h[i]) atomicAdd(&hist[i], h[i]);
}

__global__ void pick_thr_k(const unsigned* __restrict__ hist, unsigned* __restrict__ thr) {
  if (threadIdx.x == 0 && blockIdx.x == 0) {
    unsigned cum = 0;
    int b = 4095;
    for (; b >= 0; --b) { cum += hist[b]; if (cum >= (unsigned)KNUM) break; }
    if (b < 0) b = 0;
    thr[0] = ((unsigned)b) << 20;
  }
}

__global__ void nms_collect_k(const float* __restrict__ lo, const float* __restrict__ cu,
                              const float* __restrict__ hi, int H, int W,
                              const unsigned* __restrict__ thr, u64* __restrict__ buf,
                              unsigned* __restrict__ cnt, int lvl) {
  const int x = BORDER + blockIdx.x * blockDim.x + threadIdx.x;
  const int y = BORDER + blockIdx.y;
  if (x >= W - BORDER) return;
  const float c = cu[(size_t)y * W + x];
  if (!(c > 0.f)) return;
  const unsigned bits = __float_as_uint(c);
  if (bits < *thr) return;
  if (!is_peak(lo, cu, hi, W, y, x, c)) return;
  unsigned idx = atomicAdd(cnt, 1u);
  if (idx < 8192u)
    buf[idx] = ((u64)bits << 32) |
               (u64)(((unsigned)lvl << 22) | ((unsigned)y << 11) | (unsigned)x);
}

// Per-level: bitonic sort (descending) of up to 8192 candidates in LDS, emit top 2000.
__global__ __launch_bounds__(1024) void level_topk_k(const u64* __restrict__ buf,
                                                     const unsigned* __restrict__ cnt,
                                                     u64* __restrict__ out, int lvl) {
  __shared__ u64 s[8192];
  unsigned n = *cnt; if (n > 8192u) n = 8192u;
  for (int i = threadIdx.x; i < 8192; i += 1024) s[i] = (i < (int)n) ? buf[i] : 0ull;
  __syncthreads();
  for (int k = 2; k <= 8192; k <<= 1)
    for (int j = k >> 1; j > 0; j >>= 1) {
      for (int i = threadIdx.x; i < 8192; i += 1024) {
        int ixj = i ^ j;
        if (ixj > i) {
          u64 a = s[i], b = s[ixj];
          if (((i & k) == 0) ? (a < b) : (a > b)) { s[i] = b; s[ixj] = a; }
        }
      }
      __syncthreads();
    }
  // Padded entries mimic the reference's -inf rows: value-bits 0, center (0,0), level kept.
  for (int i = threadIdx.x; i < KNUM; i += 1024)
    out[i] = s[i] ? s[i] : (u64)((unsigned)lvl << 22);
}

// One step of a global descending bitonic sort over 65536 packed candidates.
__global__ void gsort_step_k(u64* __restrict__ d, int j, int k) {
  int i = blockIdx.x * blockDim.x + threadIdx.x;
  int ixj = i ^ j;
  if (ixj > i) {
    u64 a = d[i], b = d[ixj];
    if (((i & k) == 0) ? (a < b) : (a > b)) { d[i] = b; d[ixj] = a; }
  }
}

__global__ void write_out_k(const u64* __restrict__ sorted, float* __restrict__ lafs,
                            float* __restrict__ resp, Info info) {
  int i = blockIdx.x * blockDim.x + threadIdx.x;
  if (i >= KNUM) return;
  u64 e = sorted[i];
  unsigned bits = (unsigned)(e >> 32);
  unsigned p = (unsigned)e;
  int lvl = (p >> 22) & 31;
  int yy = (p >> 11) & 2047, xx = p & 2047;
  int oc = lvl / 6, l = lvl % 6 + 1;
  float pd = (float)(1 << oc);
  float s = info.sigma[l] * pd;
  lafs[i * 6 + 0] = 3.0f * s; lafs[i * 6 + 1] = 0.f;      lafs[i * 6 + 2] = xx * pd;
  lafs[i * 6 + 3] = 0.f;      lafs[i * 6 + 4] = 3.0f * s; lafs[i * 6 + 5] = yy * pd;
  resp[i] = bits ? __uint_as_float(bits) : -INFINITY;
}

__global__ __launch_bounds__(256) void patches_k(const u64* __restrict__ sorted,
                                                 const float* __restrict__ pyr,
                                                 Info info, float* __restrict__ outp) {
  const int kp = blockIdx.x;
  const u64 e = sorted[kp];
  const unsigned p = (unsigned)e;
  const int lvl = (p >> 22) & 31;
  const int yy = (p >> 11) & 2047;
  const int xx = p & 2047;
  const int oc = lvl / 6;
  const int l = lvl % 6 + 1;
  const float* img = pyr + info.offF[oc] + (u64)l * info.px[oc];
  const int H = info.H[oc], W = info.W[oc];
  const float s3 = 3.0f * info.sigma[l];  // (MRSIZE*sigma*pd)*grid / pd = 3*sigma*grid
  for (int t = threadIdx.x; t < PS * PS; t += blockDim.x) {
    const int py = t >> 5, px = t & 31;
    const float gu = ((px + 0.5f) / (float)PS) * 2.f - 1.f;
    const float gv = ((py + 0.5f) / (float)PS) * 2.f - 1.f;
    float xs = s3 * gu + (float)xx;
    float ys = s3 * gv + (float)yy;
    xs = fminf(fmaxf(xs, 0.f), (float)W - 1.001f);
    ys = fminf(fmaxf(ys, 0.f), (float)H - 1.001f);
    const int x0 = (int)floorf(xs), y0 = (int)floorf(ys);
    const int x1 = min(x0 + 1, W - 1), y1 = min(y0 + 1, H - 1);
    const float wx = xs - (float)x0, wy = ys - (float)y0;
    const float v00 = img[(size_t)y0 * W + x0], v01 = img[(size_t)y0 * W + x1];
    const float v10 = img[(size_t)y1 * W + x0], v11 = img[(size_t)y1 * W + x1];
    outp[(size_t)kp * PS * PS + t] =
        (v00 * (1.f - wx) + v01 * wx) * (1.f - wy) + (v10 * (1.f - wx) + v11 * wx) * wy;
  }
}

// ---------------------------------------------------------------------------
static void make_taps(double sigma, Taps& tp) {
  int r = std::max(1, (int)std::ceil(3.0 * sigma));
  if (r < 3) r = 3;   // our sigma sequence gives r in [3,5]; clamp for template dispatch
  if (r > 5) r = 5;
  double v[16], sum = 0.0;
  for (int i = -r; i <= r; ++i) { double e = std::exp(-0.5 * (i / sigma) * (i / sigma)); v[i + r] = e; sum += e; }
  for (int i = 0; i < 16; ++i) tp.t[i] = (i < 2 * r + 1) ? (float)(v[i] / sum) : 0.f;
  tp.r = r;
}

extern "C" void kernel_launch(void* const* d_in, const int* in_sizes, int n_in,
                              void* d_out, int out_size, void* d_ws, size_t ws_size,
                              hipStream_t stream) {
  (void)in_sizes; (void)n_in; (void)out_size; (void)ws_size;
  const float* x = (const float*)d_in[0];
  const int Ho[3] = {1536, 768, 384};
  const int Wo[3] = {2048, 1024, 512};
  const size_t px[3] = {1536ull * 2048ull, 768ull * 1024ull, 384ull * 512ull};

  const double kk = std::pow(2.0, 1.0 / (NLEVELS - 2));
  double sig[NLEVELS];
  for (int i = 0; i < NLEVELS; ++i) sig[i] = 1.6 * std::pow(kk, i);
  Taps tapsBase;
  make_taps(std::sqrt(std::max(1.6 * 1.6 - 0.25, 1e-8)), tapsBase);
  Taps tapsInc[NLEVELS];
  for (int i = 1; i < NLEVELS; ++i)
    make_taps(std::sqrt(sig[i] * sig[i] - sig[i - 1] * sig[i - 1]), tapsInc[i]);

  Info info;
  info.offF[0] = 0; info.offF[1] = 8 * px[0]; info.offF[2] = 8 * (px[0] + px[1]);
  for (int o = 0; o < 3; ++o) { info.px[o] = px[o]; info.H[o] = Ho[o]; info.W[o] = Wo[o]; }
  for (int l = 0; l < NLEVELS; ++l) info.sigma[l] = (float)sig[l];

  // workspace carve (256B aligned)
  char* base = (char*)d_ws;
  size_t off = 0;
  auto carve = [&](size_t bytes) -> char* {
    off = (off + 255) & ~(size_t)255;
    char* pp = base + off; off += bytes; return pp;
  };
  float* pyr     = (float*)carve(8 * (px[0] + px[1] + px[2]) * sizeof(float));
  float* resp    = (float*)carve(8 * (px[0] + px[1] + px[2]) * sizeof(float));
  float* tmp     = (float*)carve(px[0] * sizeof(float));
  u64*   candAll = (u64*)carve(65536 * sizeof(u64));
  u64*   lvlbuf  = (u64*)carve(8192 * sizeof(u64));
  unsigned* hist = (unsigned*)carve(4096 * sizeof(unsigned));
  unsigned* cnt  = (unsigned*)carve(sizeof(unsigned));
  unsigned* thr  = (unsigned*)carve(sizeof(unsigned));

  auto PYR  = [&](int oc, int l) { return pyr  + info.offF[oc] + (size_t)l * px[oc]; };
  auto RESP = [&](int oc, int l) { return resp + info.offF[oc] + (size_t)l * px[oc]; };
  auto blur = [&](const float* s, float* d, int oc, const Taps& tp) {
    int tiles = ((Wo[oc] + 15) / 16) * ((Ho[oc] + 15) / 16);
    int blocks = (tiles + 7) / 8;   // 8 waves (tiles) per 256-thread block
    switch (tp.r) {
      case 3:
        row_blur_wmma<3><<<blocks, 256, 0, stream>>>(s, tmp, Ho[oc], Wo[oc], tp);
        col_blur_wmma<3><<<blocks, 256, 0, stream>>>(tmp, d, Ho[oc], Wo[oc], tp);
        break;
      case 4:
        row_blur_wmma<4><<<blocks, 256, 0, stream>>>(s, tmp, Ho[oc], Wo[oc], tp);
        col_blur_wmma<4><<<blocks, 256, 0, stream>>>(tmp, d, Ho[oc], Wo[oc], tp);
        break;
      default:
        row_blur_wmma<5><<<blocks, 256, 0, stream>>>(s, tmp, Ho[oc], Wo[oc], tp);
        col_blur_wmma<5><<<blocks, 256, 0, stream>>>(tmp, d, Ho[oc], Wo[oc], tp);
        break;
    }
  };

  // 1) Gaussian pyramid
  blur(x, PYR(0, 0), 0, tapsBase);
  for (int oc = 0; oc < NOCT; ++oc) {
    if (oc > 0) {
      dim3 g((Wo[oc] + 255) / 256, Ho[oc]);
      downsample_k<<<g, 256, 0, stream>>>(PYR(oc - 1, NLEVELS - 2), PYR(oc, 0),
                                          Ho[oc], Wo[oc], Wo[oc - 1]);
    }
    for (int l = 1; l < NLEVELS; ++l) blur(PYR(oc, l - 1), PYR(oc, l), oc, tapsInc[l]);
  }

  // 2) Hessian responses
  for (int oc = 0; oc < NOCT; ++oc)
    for (int l = 0; l < NLEVELS; ++l) {
      float s4 = (float)(sig[l] * sig[l] * sig[l] * sig[l]);
      dim3 g((Wo[oc] + 255) / 256, Ho[oc]);
      hessian_k<<<g, 256, 0, stream>>>(PYR(oc, l), RESP(oc, l), Ho[oc], Wo[oc], s4);
    }

  // 3) Per-level detection + top-2000
  (void)hipMemsetAsync(candAll, 0, 65536 * sizeof(u64), stream);
  for (int oc = 0; oc < NOCT; ++oc)
    for (int l = 1; l <= NLEVELS - 2; ++l) {
      int lvl = oc * 6 + (l - 1);
      (void)hipMemsetAsync(hist, 0, 4096 * sizeof(unsigned), stream);
      (void)hipMemsetAsync(cnt, 0, sizeof(unsigned), stream);
      dim3 g((Wo[oc] - 2 * BORDER + 255) / 256, Ho[oc] - 2 * BORDER);
      nms_hist_k<<<g, 256, 0, stream>>>(RESP(oc, l - 1), RESP(oc, l), RESP(oc, l + 1),
                                        Ho[oc], Wo[oc], hist);
      pick_thr_k<<<1, 32, 0, stream>>>(hist, thr);
      nms_collect_k<<<g, 256, 0, stream>>>(RESP(oc, l - 1), RESP(oc, l), RESP(oc, l + 1),
                                           Ho[oc], Wo[oc], thr, lvlbuf, cnt, lvl);
      level_topk_k<<<1, 1024, 0, stream>>>(lvlbuf, cnt, candAll + (size_t)lvl * KNUM, lvl);
    }

  // 4) Global descending bitonic sort over 65536 (36000 live + zero padding)
  for (int k2 = 2; k2 <= 65536; k2 <<= 1)
    for (int j = k2 >> 1; j > 0; j >>= 1)
      gsort_step_k<<<256, 256, 0, stream>>>(candAll, j, k2);

  // 5) Outputs: lafs [2000,2,3] | patches [2000,1,32,32] | resp [2000]
  float* out = (float*)d_out;
  float* out_lafs  = out;
  float* out_patch = out + (size_t)KNUM * 6;
  float* out_resp  = out + (size_t)KNUM * 6 + (size_t)KNUM * PS * PS;
  write_out_k<<<(KNUM + 255) / 256, 256, 0, stream>>>(candAll, out_lafs, out_resp, info);
  patches_k<<<KNUM, 256, 0, stream>>>(candAll, pyr, info, out_patch);
}